// RBFKernel_65481071401357
// MI455X (gfx1250) — compile-verified
//
#include <hip/hip_runtime.h>

// RBF kernel: out[i,j] = exp(-GAMMA * max(||x_i||^2 + ||y_j||^2 - 2*x_i.y_j, 0))
// x: [8192,64] f32, y: [8192,64] f32, out: [8192,8192] f32
// GEMM core via V_WMMA_F32_16X16X4_F32 (wave32, CDNA5).

typedef __attribute__((ext_vector_type(2))) float v2f;
typedef __attribute__((ext_vector_type(8))) float v8f;
typedef __attribute__((ext_vector_type(4))) float f4;

#define GAMMA_F 1.0f

constexpr int NROWS = 8192;  // rows of x  (output rows)
constexpr int MROWS = 8192;  // rows of y  (output cols)
constexpr int DDIM  = 64;    // feature dim (K)

constexpr int BM  = 128;     // output rows per workgroup
constexpr int BN  = 64;      // output cols per workgroup
constexpr int LDX = 68;      // padded LDS row stride (floats): 68*4B breaks 64-bank conflicts,
                             // and 68*4 = 272 B keeps float4 (16B) alignment per row.

__global__ __launch_bounds__(256) void rbf_wmma_kernel(const float* __restrict__ x,
                                                       const float* __restrict__ y,
                                                       float* __restrict__ out) {
    __shared__ float sX[BM * LDX];   // 128 x 64 panel of x (padded)
    __shared__ float sY[BN * LDX];   //  64 x 64 panel of y (padded)
    __shared__ float sXsq[BM];
    __shared__ float sYsq[BN];

    const int tid = threadIdx.x;
    const int bM0 = blockIdx.y * BM;
    const int bN0 = blockIdx.x * BN;

    // ---- cooperative stage: global -> LDS as float4 (L2-resident inputs) ----
    {
        const f4* gx = (const f4*)(x + (size_t)bM0 * DDIM);
        // 128 rows * 16 quads = 2048 quads; 256 threads -> 8 each
#pragma unroll
        for (int i = 0; i < 8; ++i) {
            int idx = tid + i * 256;
            int row = idx >> 4;
            int q   = idx & 15;
            f4 v = gx[row * (DDIM / 4) + q];
            *(f4*)(&sX[row * LDX + q * 4]) = v;
        }
        const f4* gy = (const f4*)(y + (size_t)bN0 * DDIM);
        // 64 rows * 16 quads = 1024 quads; 256 threads -> 4 each
#pragma unroll
        for (int i = 0; i < 4; ++i) {
            int idx = tid + i * 256;
            int row = idx >> 4;
            int q   = idx & 15;
            f4 v = gy[row * (DDIM / 4) + q];
            *(f4*)(&sY[row * LDX + q * 4]) = v;
        }
    }
    __syncthreads();

    // ---- row norms (once per block) ----
    if (tid < BM) {
        float s = 0.0f;
#pragma unroll
        for (int d = 0; d < DDIM; ++d) {
            float v = sX[tid * LDX + d];
            s += v * v;
        }
        sXsq[tid] = s;
    } else if (tid < BM + BN) {
        int r = tid - BM;
        float s = 0.0f;
#pragma unroll
        for (int d = 0; d < DDIM; ++d) {
            float v = sY[r * LDX + d];
            s += v * v;
        }
        sYsq[r] = s;
    }
    __syncthreads();

    // ---- WMMA main loop: each wave computes a 16(M) x 64(N) strip ----
    const int lane  = tid & 31;
    const int wave  = tid >> 5;     // 0..7 (wave32)
    const int half  = lane >> 4;    // 0 or 1
    const int mloc  = lane & 15;    // row/col within 16-tile
    const int waveM = wave * 16;    // wave's row offset within block

    v8f acc[4];
#pragma unroll
    for (int t = 0; t < 4; ++t) acc[t] = (v8f){};

    // A-fragment layout (16x4 f32): lane L -> M = L%16; VGPR0 = K=2*(L/16), VGPR1 = K+1
    const float* aRow = &sX[(waveM + mloc) * LDX];

#pragma unroll
    for (int k = 0; k < DDIM; k += 4) {
        const int k0 = k + 2 * half;
        v2f a;
        a.x = aRow[k0];
        a.y = aRow[k0 + 1];

#pragma unroll
        for (int t = 0; t < 4; ++t) {
            const float* bRow = &sY[(t * 16 + mloc) * LDX];
            v2f b;
            b.x = bRow[k0];
            b.y = bRow[k0 + 1];
            // D = A(16x4) * B(4x16) + C   -> v_wmma_f32_16x16x4_f32
            acc[t] = __builtin_amdgcn_wmma_f32_16x16x4_f32(
                /*neg_a=*/false, a, /*neg_b=*/false, b,
                /*c_mod=*/(short)0, acc[t], /*reuse_a=*/false, /*reuse_b=*/false);
        }
    }

    // ---- epilogue: dist^2 -> exp, store ----
    // C/D layout: VGPR v, lane L -> M = v + 8*(L/16), N = L%16
#pragma unroll
    for (int t = 0; t < 4; ++t) {
        const int nloc = t * 16 + mloc;
        const float ysq = sYsq[nloc];
#pragma unroll
        for (int v = 0; v < 8; ++v) {
            const int m  = waveM + v + 8 * half;
            float d2 = sXsq[m] + ysq - 2.0f * acc[t][v];
            d2 = fmaxf(d2, 0.0f);
            const size_t oidx = (size_t)(bM0 + m) * (size_t)MROWS + (size_t)(bN0 + nloc);
            out[oidx] = __expf(-GAMMA_F * d2);
        }
    }
}

extern "C" void kernel_launch(void* const* d_in, const int* in_sizes, int n_in,
                              void* d_out, int out_size, void* d_ws, size_t ws_size,
                              hipStream_t stream) {
    (void)in_sizes; (void)n_in; (void)out_size; (void)d_ws; (void)ws_size;
    const float* x = (const float*)d_in[0];
    const float* y = (const float*)d_in[1];
    float* out = (float*)d_out;

    dim3 grid(MROWS / BN, NROWS / BM);  // (128, 64)
    dim3 block(256);
    rbf_wmma_kernel<<<grid, block, 0, stream>>>(x, y, out);
}